// DistanceLoss_31258771980827
// MI455X (gfx1250) — compile-verified
//
#include <hip/hip_runtime.h>
#include <hip/hip_bf16.h>
#include <math.h>

// Problem constants (reference: predictions [16,5,256,256], targets [16,256,256])
#define Hh 256
#define Ww 256
#define HW (Hh * Ww)
#define CMAX 8
#define ROWS 16          // row tile per block in min-plus kernel
#define BIGF 512.0f      // H + W, matches reference BIG

typedef float v2f __attribute__((ext_vector_type(2)));
typedef float v8f __attribute__((ext_vector_type(8)));

// ---------------- kernel 0: zero the per-slice max accumulators ----------------
__global__ void k_init(unsigned int* __restrict__ maxd2, int S) {
    int t = blockIdx.x * blockDim.x + threadIdx.x;
    if (t < S) maxd2[t] = 0u;   // bits of +0.0f
}

// ---------------- kernel 1: exact 1D row distance (squared) --------------------
// one thread per (slice, row): fwd scan, store; bwd scan, min, clamp, square.
__global__ void k_rowdist(const int* __restrict__ targets,
                          float* __restrict__ g2, int B, int C) {
    int t = blockIdx.x * blockDim.x + threadIdx.x;
    int total = B * C * Hh;
    if (t >= total) return;
    int i = t % Hh;
    int s = t / Hh;          // slice = b*C + c
    int c = s % C;
    int b = s / C;
    const int* trow = targets + (size_t)(b * Hh + i) * Ww;
    float* grow = g2 + (size_t)s * HW + (size_t)i * Ww;

    float cnt = BIGF;                         // init carry = BIG (reference)
    for (int j = 0; j < Ww; ++j) {
        bool nz = (trow[j] != c);             // nonzero of ~present
        cnt = nz ? cnt + 1.0f : 0.0f;
        grow[j] = cnt;                        // stash fwd distance
    }
    cnt = BIGF;
    for (int j = Ww - 1; j >= 0; --j) {
        bool nz = (trow[j] != c);
        cnt = nz ? cnt + 1.0f : 0.0f;
        float g = fminf(fminf(grow[j], cnt), BIGF);
        grow[j] = g * g;                      // store squared 1D distance
    }
}

// ---------------- kernel 2: column min-plus (tropical conv) --------------------
// block = (slice, 16-row tile); threadIdx.x = column j. Each coalesced g2 load
// feeds 16 register accumulators. Also reduces per-slice max(d^2).
__global__ void k_minplus(const float* __restrict__ g2,
                          float* __restrict__ d2,
                          unsigned int* __restrict__ maxd2) {
    const int tilesPerSlice = Hh / ROWS;
    int s    = blockIdx.x / tilesPerSlice;
    int tile = blockIdx.x % tilesPerSlice;
    int j    = threadIdx.x;                   // 0..255
    int i0   = tile * ROWS;

    const float* gs = g2 + (size_t)s * HW;

    float dmin[ROWS];
#pragma unroll
    for (int r = 0; r < ROWS; ++r) dmin[r] = 3.0e38f;

    for (int ip = 0; ip < Hh; ++ip) {
        float v = gs[(size_t)ip * Ww + j];    // coalesced across lanes
        float base = (float)(i0 - ip);
#pragma unroll
        for (int r = 0; r < ROWS; ++r) {
            float di = base + (float)r;
            dmin[r] = fminf(dmin[r], fmaf(di, di, v));
        }
    }

    float tmax = 0.0f;
    float* drow = d2 + (size_t)s * HW + (size_t)i0 * Ww + j;
#pragma unroll
    for (int r = 0; r < ROWS; ++r) {
        drow[(size_t)r * Ww] = dmin[r];
        tmax = fmaxf(tmax, dmin[r]);
    }

    // block-wide max -> per-slice atomicMax (uint bits; all values >= 0)
    __shared__ float red[256];
    red[threadIdx.x] = tmax;
    __syncthreads();
    for (int off = 128; off > 0; off >>= 1) {
        if (threadIdx.x < off) red[threadIdx.x] = fmaxf(red[threadIdx.x], red[threadIdx.x + off]);
        __syncthreads();
    }
    if (threadIdx.x == 0) atomicMax(&maxd2[s], __float_as_uint(red[0]));
}

// ---------------- kernel 3: softmax * dist * weight, block partial sums --------
__global__ void k_loss(const float* __restrict__ preds,
                       const float* __restrict__ weight,
                       const int* __restrict__ targets,
                       const float* __restrict__ d2,
                       const unsigned int* __restrict__ maxd2,
                       float* __restrict__ partials, int B, int C) {
    int idx = blockIdx.x * blockDim.x + threadIdx.x;
    int P = B * HW;
    float acc = 0.0f;
    if (idx < P) {
        int b   = idx / HW;
        int rem = idx % HW;

        float lg[CMAX], e[CMAX];
        float m = -3.0e38f;
        for (int c = 0; c < C; ++c) {
            lg[c] = preds[((size_t)(b * C + c)) * HW + rem];
            m = fmaxf(m, lg[c]);
        }
        float se = 0.0f;
        for (int c = 0; c < C; ++c) { e[c] = __expf(lg[c] - m); se += e[c]; }
        float inv = 1.0f / se;

        float wsum = 0.0f;
        for (int c = 0; c < C; ++c) wsum += weight[c];
        float winv = 1.0f / wsum;

        int tgt = targets[idx];               // targets flat is [B][H][W]
        for (int c = 0; c < C; ++c) {
            size_t o = ((size_t)(b * C + c)) * HW + rem;
            float dd = d2[o];
            float dist = (tgt == c) ? -sqrtf(__uint_as_float(maxd2[b * C + c]))
                                    : sqrtf(dd);
            acc += (e[c] * inv) * dist * (weight[c] * winv);
        }
    }
    __shared__ float red[256];
    red[threadIdx.x] = acc;
    __syncthreads();
    for (int off = 128; off > 0; off >>= 1) {
        if (threadIdx.x < off) red[threadIdx.x] += red[threadIdx.x + off];
        __syncthreads();
    }
    if (threadIdx.x == 0) partials[blockIdx.x] = red[0];
}

// ---------------- kernel 4: WMMA f32 final reduction (one wave) ----------------
// A (16x4 f32) carries 64 partials per issue, B = all-ones (layout-free since
// every element is 1.0), C accumulates. Total = column 0 of D = lane0 + lane16
// row-sums. Exact f32 math via V_WMMA_F32_16X16X4_F32.
// Main loop: unconditional 64-bit vector loads (no EXEC divergence); tail is a
// wave-uniform branch so EXEC stays all-ones for every WMMA.
__global__ void k_final(const float* __restrict__ partials, int n,
                        float* __restrict__ out, float invN) {
    int lane = threadIdx.x;                   // single wave32, EXEC all ones
    v8f c = {0.0f, 0.0f, 0.0f, 0.0f, 0.0f, 0.0f, 0.0f, 0.0f};
    v2f ones; ones.x = 1.0f; ones.y = 1.0f;

    const v2f* p2 = (const v2f*)partials;     // 8-byte aligned by ws layout
    int full = n >> 6;                        // iterations of 64 values
    for (int it = 0; it < full; ++it) {
        v2f a = p2[it * 32 + lane];           // global_load_b64, coalesced
        c = __builtin_amdgcn_wmma_f32_16x16x4_f32(
                false, a, false, ones, (short)0, c, false, false);
    }
    if (n & 63) {                             // uniform branch (scalar n)
        int base = (full << 6) + lane * 2;
        v2f a;
        a.x = (base     < n) ? partials[base]     : 0.0f;
        a.y = (base + 1 < n) ? partials[base + 1] : 0.0f;
        c = __builtin_amdgcn_wmma_f32_16x16x4_f32(
                false, a, false, ones, (short)0, c, false, false);
    }
    float s = c[0] + c[1] + c[2] + c[3] + c[4] + c[5] + c[6] + c[7];
    // D column 0: rows 0-7 live in lane 0, rows 8-15 in lane 16
    float total = __shfl(s, 0, 32) + __shfl(s, 16, 32);
    if (lane == 0) out[0] = total * invN;
}

// -------------------------------------------------------------------------------
extern "C" void kernel_launch(void* const* d_in, const int* in_sizes, int n_in,
                              void* d_out, int out_size, void* d_ws, size_t ws_size,
                              hipStream_t stream) {
    const float* preds   = (const float*)d_in[0];
    const float* weight  = (const float*)d_in[1];
    const int*   targets = (const int*)d_in[2];
    float* out = (float*)d_out;

    int C = in_sizes[1];                       // 5
    int B = in_sizes[2] / HW;                  // 16
    int S = B * C;                             // 80 slices

    // workspace layout (partials lands 8-byte aligned: 2*S*HW floats + S uints)
    float* g2 = (float*)d_ws;
    float* d2 = g2 + (size_t)S * HW;
    unsigned int* maxd2 = (unsigned int*)(d2 + (size_t)S * HW);
    float* partials = (float*)(maxd2 + S);

    int P = B * HW;
    int lossBlocks = (P + 255) / 256;          // 4096 (multiple of 64)

    k_init<<<(S + 127) / 128, 128, 0, stream>>>(maxd2, S);
    k_rowdist<<<(S * Hh + 255) / 256, 256, 0, stream>>>(targets, g2, B, C);
    k_minplus<<<S * (Hh / ROWS), 256, 0, stream>>>(g2, d2, maxd2);
    k_loss<<<lossBlocks, 256, 0, stream>>>(preds, weight, targets, d2, maxd2,
                                           partials, B, C);
    float invN = 1.0f / (float)((size_t)B * C * HW);
    k_final<<<1, 32, 0, stream>>>(partials, lossBlocks, out, invN);
}